// ExampleGNN_57767310131483
// MI455X (gfx1250) — compile-verified
//
#include <hip/hip_runtime.h>

typedef __attribute__((ext_vector_type(2))) float v2f;
typedef __attribute__((ext_vector_type(8))) float v8f;

#define GD 128  // hidden feature dim (fixed by problem)
#define GC 40   // head output classes (fixed by problem)

// ---------------- degree / normalization ----------------
__global__ __launch_bounds__(256) void gcn_deg_init(float* __restrict__ deg, int N) {
  int i = blockIdx.x * blockDim.x + threadIdx.x;
  if (i < N) deg[i] = 1.0f;  // self-loop contributes 1
}

__global__ __launch_bounds__(256) void gcn_deg_count(const int* __restrict__ dst,
                                                     float* __restrict__ deg, int E) {
  int i = blockIdx.x * blockDim.x + threadIdx.x;
  if (i < E) atomicAdd(&deg[dst[i]], 1.0f);
}

__global__ __launch_bounds__(256) void gcn_dinv(float* __restrict__ deg, int N) {
  int i = blockIdx.x * blockDim.x + threadIdx.x;
  if (i < N) deg[i] = rsqrtf(deg[i]);  // deg >= 1 always (self-loop)
}

__global__ __launch_bounds__(256) void gcn_zero(float4* __restrict__ buf, int n4) {
  int i = blockIdx.x * blockDim.x + threadIdx.x;
  if (i < n4) buf[i] = make_float4(0.f, 0.f, 0.f, 0.f);
}

// ---------------- WMMA f32 GEMM: Out[M x NCOLS] = A[M x 128] @ B[128 x NCOLS] (+bias) ----
// One wave per 16x16 output tile; K=128 -> 32 chained v_wmma_f32_16x16x4_f32.
// NCOLS is compile-time so all loads use immediate offsets (no per-iter addr math).
template <int NCOLS>
__global__ __launch_bounds__(256) void gcn_gemm_wmma(
    const float* __restrict__ A, const float* __restrict__ B,
    const float* __restrict__ bias, float* __restrict__ Out, int M)
{
  constexpr int NCT = (NCOLS + 15) / 16;        // column tiles
  constexpr bool MASK = (NCOLS % 16) != 0;      // need column masking?

  int wave = blockIdx.x * 8 + (threadIdx.x >> 5);
  int lane = threadIdx.x & 31;
  int M16 = (M + 15) >> 4;
  if (wave >= M16 * NCT) return;                // uniform per-wave exit
  int rt = wave / NCT;
  int ct = wave - rt * NCT;

  int lm = lane & 15;
  int m  = rt * 16 + lm;
  int mc = m < M ? m : M - 1;                   // clamp (loads only)
  int n  = ct * 16 + lm;
  int ncl   = (MASK && n >= NCOLS) ? (NCOLS - 1) : n;
  float msk = (MASK && n >= NCOLS) ? 0.0f : 1.0f;
  int kg = (lane >> 4) << 1;                    // 0 (lanes 0-15) or 2 (lanes 16-31)

  // element A[mc][k0+kg]   = arow[k0]      (8B-aligned pairs)
  // element B[k0+kg][ncl]  = bcol[k0*NCOLS] (immediate offsets, NCOLS constexpr)
  const float* arow = A + (size_t)mc * GD + kg;
  const float* bcol = B + (size_t)kg * NCOLS + ncl;

  v8f acc = {};
#pragma unroll
  for (int k0 = 0; k0 < GD; k0 += 4) {
    v2f a = *(const v2f*)(arow + k0);
    v2f b;
    if (MASK) {
      b.x = bcol[k0 * NCOLS] * msk;
      b.y = bcol[(k0 + 1) * NCOLS] * msk;
    } else {
      b.x = bcol[k0 * NCOLS];
      b.y = bcol[(k0 + 1) * NCOLS];
    }
    acc = __builtin_amdgcn_wmma_f32_16x16x4_f32(
        false, a, false, b, (short)0, acc, false, false);
  }

  float bv = bias ? bias[ncl] : 0.0f;
  int rbase = rt * 16 + ((lane >> 4) << 3);     // lanes 16-31 hold M = r + 8
  if (!MASK || n < NCOLS) {
    float* orow = Out + (size_t)rbase * NCOLS + n;
    int rmax = M - rbase;
#pragma unroll
    for (int r = 0; r < 8; ++r) {
      if (r < rmax) orow[r * NCOLS] = acc[r] + bv;   // immediate offsets
    }
  }
}

// ---------------- edge scatter-add: one wave per edge, float4 per lane ----------------
__global__ __launch_bounds__(256) void gcn_scatter(
    const float* __restrict__ hw, const int* __restrict__ src, const int* __restrict__ dst,
    const float* __restrict__ dinv, float* __restrict__ agg, int E, int N)
{
  int w = blockIdx.x * 8 + (threadIdx.x >> 5);
  int lane = threadIdx.x & 31;
  if (w >= E + N) return;
  int s, d;
  if (w < E) { s = src[w]; d = dst[w]; }
  else       { s = w - E; d = s; }              // self-loop edges
  float norm = dinv[s] * dinv[d];
  const float4* row = (const float4*)(hw + (size_t)s * GD);
  float4 v = row[lane];                         // 32 lanes x 4 floats = 128
  float* o = agg + (size_t)d * GD + lane * 4;
  atomicAdd(o + 0, v.x * norm);
  atomicAdd(o + 1, v.y * norm);
  atomicAdd(o + 2, v.z * norm);
  atomicAdd(o + 3, v.w * norm);
}

__global__ __launch_bounds__(256) void gcn_bias_relu(float* __restrict__ buf,
                                                     const float* __restrict__ bias, int total) {
  int i = blockIdx.x * blockDim.x + threadIdx.x;
  if (i < total) {
    float v = buf[i] + bias[i & (GD - 1)];
    buf[i] = fmaxf(v, 0.0f);
  }
}

extern "C" void kernel_launch(void* const* d_in, const int* in_sizes, int n_in,
                              void* d_out, int out_size, void* d_ws, size_t ws_size,
                              hipStream_t stream) {
  const float* x  = (const float*)d_in[0];
  const int*   ei = (const int*)  d_in[1];
  const float* W1 = (const float*)d_in[2];
  const float* b1 = (const float*)d_in[3];
  const float* W2 = (const float*)d_in[4];
  const float* b2 = (const float*)d_in[5];
  const float* Wh = (const float*)d_in[6];
  const float* bh = (const float*)d_in[7];
  float* out = (float*)d_out;

  const int N = in_sizes[0] / GD;
  const int E = in_sizes[1] / 2;
  const int* src = ei;
  const int* dst = ei + E;

  // workspace layout: dinv [N, padded], bufA [N*GD], bufB [N*GD]  (~10.3 MB)
  float* dinv = (float*)d_ws;
  float* bufA = dinv + (((size_t)N + 127) & ~(size_t)127);
  float* bufB = bufA + (size_t)N * GD;

  const int ND  = N * GD;
  const int M16 = (N + 15) / 16;

  // --- symmetric normalization ---
  gcn_deg_init<<<(N + 255) / 256, 256, 0, stream>>>(dinv, N);
  gcn_deg_count<<<(E + 255) / 256, 256, 0, stream>>>(dst, dinv, E);
  gcn_dinv<<<(N + 255) / 256, 256, 0, stream>>>(dinv, N);

  const int gemm_blocks = (M16 * (GD / 16) + 7) / 8;
  const int head_blocks = (M16 * ((GC + 15) / 16) + 7) / 8;
  const int scat_blocks = (E + N + 7) / 8;
  const int zero_blocks = (ND / 4 + 255) / 256;
  const int ew_blocks   = (ND + 255) / 256;

  // --- layer 1: hw = x@W1 ; agg ; +b1 ; relu ---
  gcn_gemm_wmma<GD><<<gemm_blocks, 256, 0, stream>>>(x, W1, nullptr, bufA, N);
  gcn_zero<<<zero_blocks, 256, 0, stream>>>((float4*)bufB, ND / 4);
  gcn_scatter<<<scat_blocks, 256, 0, stream>>>(bufA, src, dst, dinv, bufB, E, N);
  gcn_bias_relu<<<ew_blocks, 256, 0, stream>>>(bufB, b1, ND);

  // --- layer 2 ---
  gcn_gemm_wmma<GD><<<gemm_blocks, 256, 0, stream>>>(bufB, W2, nullptr, bufA, N);
  gcn_zero<<<zero_blocks, 256, 0, stream>>>((float4*)bufB, ND / 4);
  gcn_scatter<<<scat_blocks, 256, 0, stream>>>(bufA, src, dst, dinv, bufB, E, N);
  gcn_bias_relu<<<ew_blocks, 256, 0, stream>>>(bufB, b2, ND);

  // --- head: out = h2 @ Wh + bh (NCOLS=40 -> 3 masked col tiles) ---
  gcn_gemm_wmma<GC><<<head_blocks, 256, 0, stream>>>(bufB, Wh, bh, out, N);
}